// CanonicalGraphAdapter_1906965480066
// MI455X (gfx1250) — compile-verified
//
#include <hip/hip_runtime.h>
#include <math.h>

typedef float v2f __attribute__((ext_vector_type(2)));
typedef float v8f __attribute__((ext_vector_type(8)));

#define N_DIM 11
#define F_DIM 16
#define H_DIM 32
#define LN_EPS 1e-5f
#define CHUNK 8   // pairs per wave; 8 | T so one chunk never straddles b

// D(16x16,f32) = A(16x4,f32) * B(4x16,f32) + C   -- fp32 matrix pipe op
__device__ __forceinline__ v8f wmma_f32(v2f a, v2f b, v8f c) {
  return __builtin_amdgcn_wmma_f32_16x16x4_f32(false, a, false, b,
                                               (short)0, c, false, false);
}

// half-wave swap (lanes l <-> l^16) via ds_swizzle SWAPX16 (xor=0x10,and=0x1f)
__device__ __forceinline__ float swap16(float v) {
  return __builtin_bit_cast(
      float, __builtin_amdgcn_ds_swizzle(__builtin_bit_cast(int, v), 0x401f));
}

// exact-GELU via A&S 7.1.26 erf approximation (|err| < 1.5e-7), branch-free:
// gelu(x) = x * Phi(x);  Phi = (x>=0) ? 1 - p(t)*e  :  p(t)*e
// t = 1/(1 + 0.3275911*|x|/sqrt2), e = exp2(-(x^2/2)*log2e); coeffs pre-halved
__device__ __forceinline__ float gelu_fast(float xv) {
  const float z = fabsf(xv) * 0.70710678118654752f;
  const float t = __builtin_amdgcn_rcpf(fmaf(z, 0.3275911f, 1.0f));
  float p = fmaf(0.5307027145f, t, -0.7265760135f);
  p = fmaf(p, t, 0.7107068705f);
  p = fmaf(p, t, -0.142248368f);
  p = fmaf(p, t, 0.127414796f);
  p = p * t;
  const float e   = __builtin_amdgcn_exp2f(z * z * -1.4426950408889634f);
  const float hpe = p * e;                       // = 0.5*(1-erf(z)), z>=0
  const float phi = (xv >= 0.0f) ? (1.0f - hpe) : hpe;
  return xv * phi;
}

__global__ __launch_bounds__(256) void fused_proj_ln_gelu_mlp(
    const float* __restrict__ x, const int* __restrict__ lab_idx,
    const float* __restrict__ projection, const float* __restrict__ bias,
    const float* __restrict__ w1, const float* __restrict__ b1,
    const float* __restrict__ ln_g, const float* __restrict__ ln_b,
    const float* __restrict__ w2, const float* __restrict__ b2,
    float* __restrict__ out, int P, int T) {
  __shared__ float smem[8 * 544];
  const int lane = threadIdx.x & 31;
  const int wib  = threadIdx.x >> 5;
  const int wave = blockIdx.x * (blockDim.x >> 5) + wib;
  // wave-uniform; readfirstlane -> SGPR base addresses, saddr-form VMEM
  const int p0 = __builtin_amdgcn_readfirstlane(wave * CHUNK);
  if (p0 >= P) return;                 // no block-wide barriers used: safe
  const int fcol = lane & 15;          // N-column / A-row M owned by lane
  const int hi   = lane >> 4;          // half-wave select
  float* lds = smem + wib * 544;

  // ---- constant weights in WMMA B-operand layout (once per wave) ----
  v2f w1b[2][4];
  #pragma unroll
  for (int jt = 0; jt < 2; ++jt)
    #pragma unroll
    for (int k = 0; k < 4; ++k) {
      w1b[jt][k].x = w1[(4 * k + 2 * hi) * H_DIM + jt * 16 + fcol];
      w1b[jt][k].y = w1[(4 * k + 1 + 2 * hi) * H_DIM + jt * 16 + fcol];
    }
  v2f w2b[8];
  #pragma unroll
  for (int k = 0; k < 8; ++k) {
    w2b[k].x = w2[(4 * k + 2 * hi) * F_DIM + fcol];
    w2b[k].y = w2[(4 * k + 1 + 2 * hi) * F_DIM + fcol];
  }
  // per-lane LN gamma/beta for the 16 h-columns this lane owns in A-layout
  v2f gamk[8], betk[8];
  #pragma unroll
  for (int k = 0; k < 8; ++k) {
    const int j0 = 4 * k + 2 * hi;
    gamk[k].x = ln_g[j0];     gamk[k].y = ln_g[j0 + 1];
    betk[k].x = ln_b[j0];     betk[k].y = ln_b[j0 + 1];
  }
  // bias vectors folded into WMMA C operands (broadcast per lane)
  v8f cB1a, cB1b, cB2;
  {
    const float b1lo = b1[fcol], b1hh = b1[16 + fcol], b2v = b2[fcol];
    #pragma unroll
    for (int v = 0; v < 8; ++v) { cB1a[v] = b1lo; cB1b[v] = b1hh; cB2[v] = b2v; }
  }

  // ---- per-chunk (fixed b / lab): projection A-operand + bias, masked ----
  const int  bidx = p0 / T;
  const int  lab  = lab_idx[bidx];
  const float* Wp = projection + (size_t)lab * (N_DIM * N_DIM); // [n,m]
  const float* Bb = bias + (size_t)lab * (N_DIM * F_DIM);       // [m,f]
  const int  mclmp = (fcol < N_DIM) ? fcol : 0;

  v2f aW[4];
  int offx0[4], offx1[4];
  #pragma unroll
  for (int k = 0; k < 4; ++k) {
    const int n0 = 4 * k + 2 * hi, n1 = n0 + 1;
    const int c0 = (n0 < N_DIM) ? n0 : 0, c1 = (n1 < N_DIM) ? n1 : 0;
    const float m0 = (fcol < N_DIM && n0 < N_DIM) ? 1.0f : 0.0f;
    const float m1 = (fcol < N_DIM && n1 < N_DIM) ? 1.0f : 0.0f;
    aW[k].x = m0 * Wp[c0 * N_DIM + mclmp];   // Wt[m,n] = W[n,m], zero-padded
    aW[k].y = m1 * Wp[c1 * N_DIM + mclmp];
    offx0[k] = c0 * F_DIM + fcol;            // clamped: A zeros kill padding
    offx1[k] = c1 * F_DIM + fcol;
  }
  v8f cBias;                                  // per-lab bias as stage-1 C tile
  #pragma unroll
  for (int v = 0; v < 8; ++v) {
    const int m = v + 8 * hi;
    cBias[v] = ((m < N_DIM) ? 1.0f : 0.0f) *
               Bb[((m < N_DIM) ? m : 0) * F_DIM + fcol];
  }

  const float* xp = x   + (size_t)p0 * (N_DIM * F_DIM);
  float*       yp = out + (size_t)p0 * (N_DIM * F_DIM);
  const int iters = (P - p0 < CHUNK) ? (P - p0) : CHUNK;

  for (int q = 0; q < iters; ++q, xp += N_DIM * F_DIM, yp += N_DIM * F_DIM) {
    // ---- stage 1: out = Wt * x + bias  (bias rides in C operand) ----
    float xv0[4], xv1[4];
    #pragma unroll
    for (int k = 0; k < 4; ++k) {
      xv0[k] = __builtin_nontemporal_load(xp + offx0[k]);
      xv1[k] = __builtin_nontemporal_load(xp + offx1[k]);
    }
    v8f acc = cBias;
    #pragma unroll
    for (int k = 0; k < 4; ++k) {
      v2f bb; bb.x = xv0[k]; bb.y = xv1[k];
      acc = wmma_f32(aW[k], bb, acc);
    }
    // park D-tile in LDS, stride 17, for A-layout reload
    #pragma unroll
    for (int v = 0; v < 8; ++v)
      lds[(v + 8 * hi) * 17 + fcol] = acc[v];
    asm volatile("s_wait_dscnt 0x0" ::: "memory");

    // ---- stage 2: h = out * w1 + b1  (b1 rides in C operands) ----
    v8f h0 = cB1a, h1 = cB1b;
    #pragma unroll
    for (int k = 0; k < 4; ++k) {
      v2f a;
      a.x = lds[fcol * 17 + 4 * k + 2 * hi];
      a.y = lds[fcol * 17 + 4 * k + 1 + 2 * hi];
      h0 = wmma_f32(a, w1b[0][k], h0);
      h1 = wmma_f32(a, w1b[1][k], h1);
    }
    // park raw h in LDS (stride 33) for A-layout reload
    #pragma unroll
    for (int v = 0; v < 8; ++v) {
      const int m = v + 8 * hi;
      lds[m * 33 + fcol]      = h0[v];
      lds[m * 33 + 16 + fcol] = h1[v];
    }
    asm volatile("s_wait_dscnt 0x0" ::: "memory");

    // ---- stage 3: reload h as A-slices; LN(32)+GELU fully in-register ----
    v2f hs[8];
    #pragma unroll
    for (int k = 0; k < 8; ++k) {
      hs[k].x = lds[fcol * 33 + 4 * k + 2 * hi];
      hs[k].y = lds[fcol * 33 + 4 * k + 1 + 2 * hi];
    }
    // each lane owns 16 of row fcol's 32 columns: in-lane partial reduction
    float s = 0.0f, qq = 0.0f;
    #pragma unroll
    for (int k = 0; k < 8; ++k) {
      s  += hs[k].x + hs[k].y;
      qq += hs[k].x * hs[k].x + hs[k].y * hs[k].y;
    }
    // one half-wave swap completes the 32-wide row reduction
    s  += swap16(s);
    qq += swap16(qq);
    const float mu   = s * (1.0f / 32.0f);
    const float var  = qq * (1.0f / 32.0f) - mu * mu;
    const float rstd = rsqrtf(var + LN_EPS);
    #pragma unroll
    for (int k = 0; k < 8; ++k) {
      const float a0 = (hs[k].x - mu) * rstd * gamk[k].x + betk[k].x;
      const float a1 = (hs[k].y - mu) * rstd * gamk[k].y + betk[k].y;
      hs[k].x = gelu_fast(a0);
      hs[k].y = gelu_fast(a1);
    }

    // ---- stage 4: y = h * w2 + b2  (b2 rides in C operand) ----
    v8f y = cB2;
    #pragma unroll
    for (int k = 0; k < 8; ++k)
      y = wmma_f32(hs[k], w2b[k], y);
    // rows m = v+8*hi; v<3 valid for both halves, v>=3 only for lo half
    #pragma unroll
    for (int v = 0; v < 3; ++v)
      __builtin_nontemporal_store(y[v], yp + (v + 8 * hi) * F_DIM + fcol);
    if (hi == 0) {                       // single exec transition for tail
      #pragma unroll
      for (int v = 3; v < 8; ++v)
        __builtin_nontemporal_store(y[v], yp + v * F_DIM + fcol);
    }
  }
}

extern "C" void kernel_launch(void* const* d_in, const int* in_sizes, int n_in,
                              void* d_out, int out_size, void* d_ws, size_t ws_size,
                              hipStream_t stream) {
  (void)n_in; (void)out_size; (void)d_ws; (void)ws_size;
  const float* x     = (const float*)d_in[0];
  const int*   lab   = (const int*)d_in[1];
  const float* proj  = (const float*)d_in[2];
  const float* bias  = (const float*)d_in[3];
  const float* w1    = (const float*)d_in[4];
  const float* b1    = (const float*)d_in[5];
  const float* ln_g  = (const float*)d_in[6];
  const float* ln_b  = (const float*)d_in[7];
  const float* w2    = (const float*)d_in[8];
  const float* b2    = (const float*)d_in[9];
  float* out = (float*)d_out;

  const int P = in_sizes[0] / (N_DIM * F_DIM);  // B*T token groups
  const int B = in_sizes[1];
  const int T = P / B;
  const int totalWaves = (P + CHUNK - 1) / CHUNK;      // one chunk per wave
  const int blocks = (totalWaves + 7) / 8;             // 8 waves per block
  fused_proj_ln_gelu_mlp<<<blocks, 256, 0, stream>>>(
      x, lab, proj, bias, w1, b1, ln_g, ln_b, w2, b2, out, P, T);
}